// TextFilterCosine_29695403884955
// MI455X (gfx1250) — compile-verified
//
#include <hip/hip_runtime.h>
#include <math.h>

typedef float v2f __attribute__((ext_vector_type(2)));
typedef float v8f __attribute__((ext_vector_type(8)));

#define NIMG 2880
#define DIM  4096
#define NTXT 4096
#define TOPK 576
#define EPSV 1e-8f
#define ROWCHUNK 64
#define NCHUNK (NIMG / ROWCHUNK)   /* 45 */

// ---------------------------------------------------------------------------
// Kernel 1a: inverse L2 norm of each image row.  2880 blocks x 256 threads.
// ---------------------------------------------------------------------------
__global__ __launch_bounds__(256)
void k_img_norms(const float* __restrict__ img, float* __restrict__ inv_norm) {
  __shared__ float red[256];
  const int row = blockIdx.x;
  const float4* p = (const float4*)(img + (size_t)row * DIM);
  float acc = 0.f;
#pragma unroll
  for (int i = 0; i < 4; ++i) {                 // 256 thr * 4 * float4 = 4096
    float4 v = p[threadIdx.x + i * 256];
    acc += v.x * v.x + v.y * v.y + v.z * v.z + v.w * v.w;
  }
  red[threadIdx.x] = acc;
  __syncthreads();
  for (int s = 128; s > 0; s >>= 1) {
    if ((int)threadIdx.x < s) red[threadIdx.x] += red[threadIdx.x + s];
    __syncthreads();
  }
  if (threadIdx.x == 0)
    inv_norm[row] = 1.0f / fmaxf(sqrtf(red[0]), EPSV);
}

// ---------------------------------------------------------------------------
// Kernel 1b: partial column sums of normalized image rows.
// grid = (DIM/256, NCHUNK); partial[chunk][j] = sum over 64 rows.
// ---------------------------------------------------------------------------
__global__ __launch_bounds__(256)
void k_img_partial(const float* __restrict__ img, const float* __restrict__ inv_norm,
                   float* __restrict__ partial) {
  const int j    = blockIdx.x * 256 + threadIdx.x;
  const int r0   = blockIdx.y * ROWCHUNK;
  float acc = 0.f;
#pragma unroll 4
  for (int i = 0; i < ROWCHUNK; ++i) {
    acc += img[(size_t)(r0 + i) * DIM + j] * inv_norm[r0 + i];
  }
  partial[(size_t)blockIdx.y * DIM + j] = acc;
}

// ---------------------------------------------------------------------------
// Kernel 1c: reduce partials (fixed order) and scale by 1/NIMG -> m[j].
// ---------------------------------------------------------------------------
__global__ __launch_bounds__(256)
void k_img_mean(const float* __restrict__ partial, float* __restrict__ m) {
  const int j = blockIdx.x * 256 + threadIdx.x;
  float acc = 0.f;
  for (int c = 0; c < NCHUNK; ++c) acc += partial[(size_t)c * DIM + j];
  m[j] = acc * (1.0f / (float)NIMG);
}

// ---------------------------------------------------------------------------
// Kernel 2: scores via WMMA GEMV.  One block (8 waves) per 16 text rows.
//   scores[t] = mask[t] ? (text[t] . m) / max(||text[t]||, eps) : -inf
// A tile: lane L holds row (base + L%16); VGPR0/1 hold K = 2*(L/16) + {0,1}.
// B tile: m broadcast to all 16 columns -> per-lane b64 of m[k0 + 2*(L/16)].
// Every column of D carries the row dot; lane0/lane16 read rows 0-7 / 8-15.
// ---------------------------------------------------------------------------
__global__ __launch_bounds__(256)
void k_scores(const float* __restrict__ text, const float* __restrict__ m,
              const int* __restrict__ mask, float* __restrict__ scores) {
  __shared__ float lds_dot[8][16];
  __shared__ float lds_ss[8 * 32];

  const int tid  = threadIdx.x;
  const int wave = tid >> 5;           // 0..7, owns K range [wave*512, +512)
  const int lane = tid & 31;
  const int mrow = lane & 15;
  const int half = lane >> 4;
  const int base = blockIdx.x * 16;    // 256 blocks cover 4096 text rows

  const float* rowp = text + (size_t)(base + mrow) * DIM;
  const int kbeg = wave * (DIM / 8);

  v8f  c = {};
  float sumsq = 0.f;
#pragma unroll 4
  for (int k0 = kbeg; k0 < kbeg + (DIM / 8); k0 += 4) {
    v2f a2 = *(const v2f*)(rowp + k0 + 2 * half);
    v2f b2 = *(const v2f*)(m    + k0 + 2 * half);
    c = __builtin_amdgcn_wmma_f32_16x16x4_f32(
        /*neg_a=*/false, a2, /*neg_b=*/false, b2,
        /*c_mod=*/(short)0, c, /*reuse_a=*/false, /*reuse_b=*/false);
    sumsq += a2.x * a2.x + a2.y * a2.y;
  }

  // D layout: VGPR r, lanes 0-15 -> M=r (any N); lanes 16-31 -> M=r+8.
  if (lane == 0) {
#pragma unroll
    for (int r = 0; r < 8; ++r) lds_dot[wave][r] = c[r];
  }
  if (lane == 16) {
#pragma unroll
    for (int r = 0; r < 8; ++r) lds_dot[wave][8 + r] = c[r];
  }
  lds_ss[wave * 32 + lane] = sumsq;    // lane & lane+16 are same row's halves
  __syncthreads();

  if (tid < 16) {
    float dot = 0.f, ss = 0.f;
    for (int w = 0; w < 8; ++w) {      // fixed order -> deterministic
      dot += lds_dot[w][tid];
      ss  += lds_ss[w * 32 + tid] + lds_ss[w * 32 + 16 + tid];
    }
    const int t = base + tid;
    float s = dot / fmaxf(sqrtf(ss), EPSV);
    scores[t] = (mask[t] != 0) ? s : -__builtin_inff();
  }
}

// ---------------------------------------------------------------------------
// Kernel 3a: exact descending rank with stable tie-break (lowest index first).
// rank is a bijection onto [0, NTXT); selected[rank] = t for rank < TOPK.
// 4 blocks x 1024 threads, scores staged in LDS.
// ---------------------------------------------------------------------------
__global__ __launch_bounds__(1024)
void k_rank(const float* __restrict__ scores, int* __restrict__ selected) {
  __shared__ float s[NTXT];
#pragma unroll
  for (int i = 0; i < NTXT / 1024; ++i)
    s[threadIdx.x + i * 1024] = scores[threadIdx.x + i * 1024];
  __syncthreads();

  const int t  = blockIdx.x * 1024 + threadIdx.x;
  const float st = s[t];
  int rank = 0;
  for (int j = 0; j < NTXT; ++j) {
    float sj = s[j];
    rank += (sj > st) || ((sj == st) && (j < t));
  }
  if (rank < TOPK) selected[rank] = t;
}

// ---------------------------------------------------------------------------
// Kernel 3b: gather selected text rows to output.  576 blocks x 256 threads.
// ---------------------------------------------------------------------------
__global__ __launch_bounds__(256)
void k_gather(const float* __restrict__ text, const int* __restrict__ selected,
              float* __restrict__ out) {
  const int row = selected[blockIdx.x];
  const float4* src = (const float4*)(text + (size_t)row * DIM);
  float4* dst = (float4*)(out + (size_t)blockIdx.x * DIM);
#pragma unroll
  for (int i = 0; i < 4; ++i)
    dst[threadIdx.x + i * 256] = src[threadIdx.x + i * 256];
}

// ---------------------------------------------------------------------------
extern "C" void kernel_launch(void* const* d_in, const int* in_sizes, int n_in,
                              void* d_out, int out_size, void* d_ws, size_t ws_size,
                              hipStream_t stream) {
  const float* img  = (const float*)d_in[0];   // [2880, 4096]
  const float* text = (const float*)d_in[1];   // [4096, 4096]
  const int*   mask = (const int*)d_in[2];     // [4096]
  float* out = (float*)d_out;                  // [576, 4096]

  char* ws = (char*)d_ws;
  float* inv_norm = (float*)(ws);                         // 2880 f
  float* partial  = (float*)(ws + 16384);                 // 45*4096 f
  float* m        = (float*)(ws + 16384 + NCHUNK * DIM * 4);
  float* scores   = (float*)(ws + 16384 + (NCHUNK + 1) * DIM * 4);
  int*   selected = (int*)  (ws + 16384 + (NCHUNK + 2) * DIM * 4);

  k_img_norms <<<NIMG, 256, 0, stream>>>(img, inv_norm);
  {
    dim3 g(DIM / 256, NCHUNK);
    k_img_partial<<<g, 256, 0, stream>>>(img, inv_norm, partial);
  }
  k_img_mean  <<<DIM / 256, 256, 0, stream>>>(partial, m);
  k_scores    <<<NTXT / 16, 256, 0, stream>>>(text, m, mask, scores);
  k_rank      <<<NTXT / 1024, 1024, 0, stream>>>(scores, selected);
  k_gather    <<<TOPK, 256, 0, stream>>>(text, selected, out);
}